// MultiSpectralAttentionLayer_77197742178885
// MI455X (gfx1250) — compile-verified
//
#include <hip/hip_runtime.h>
#include <hip/hip_bf16.h>
#include <math.h>

#define CCH   192
#define HWSZ  3136      // 56*56
#define NBAT  64
#define DCTN  7
#define EPSV  1e-5f

typedef float v2f __attribute__((ext_vector_type(2)));
typedef float v4f __attribute__((ext_vector_type(4)));
typedef float v8f __attribute__((ext_vector_type(8)));

__constant__ int cLOWX[32] = {0,0,1,1,0,2,2,1,2,0,3,4,0,1,3,0,1,2,3,4,5,0,1,2,3,4,5,6,1,2,3,4};
__constant__ int cLOWY[32] = {0,1,0,1,2,0,1,2,2,3,0,0,4,3,1,5,4,3,2,1,0,6,5,4,3,2,1,0,6,5,4,3};

// DCT basis coefficient: cos(pi*f*(t+0.5)/7)/sqrt(7) * (f==0 ? 1 : sqrt(2))
__device__ __forceinline__ float bfc(int f, int t) {
    float r = cosf(3.14159265358979323846f * (float)f * ((float)t + 0.5f) * (1.0f / 7.0f))
              * 0.3779644730092272f;   // 1/sqrt(7)
    return f ? r * 1.4142135623730951f : r;
}

// ---------------------------------------------------------------------------
// K1: adaptive avg-pool 56x56 -> 7x7 for every (n,c), atomic-free.
// One block per (n,c) row; thread t = (h, wblock) sums 8 contiguous floats via
// two b128 loads; 49-way LDS tree reduction over the 8 rows of each h-block.
// Default (RT) temporal hint on purpose: this pass warms the 192MB L2 with x
// so the final elementwise pass re-reads x at L2 speed.
// ---------------------------------------------------------------------------
__global__ void pool_kernel(const float* __restrict__ x, float* __restrict__ xp) {
    __shared__ float part[392];                      // 56 rows * 7 w-blocks
    const int nc = blockIdx.x;
    const int t  = threadIdx.x;                      // blockDim = 392
    if (t < 392) {
        const int h  = t / 7;
        const int wb = t - 7 * h;
        const v4f* p = (const v4f*)(x + (size_t)nc * HWSZ + h * 56 + wb * 8);
        v4f a = p[0], b = p[1];
        part[t] = (a.x + a.y + a.z + a.w) + (b.x + b.y + b.z + b.w);
    }
    __syncthreads();
    if (t < 49) {
        const int hb = t / 7;
        const int wb = t - 7 * hb;
        float s = 0.0f;
#pragma unroll
        for (int r = 0; r < 8; ++r) s += part[(hb * 8 + r) * 7 + wb];
        xp[(size_t)nc * 49 + t] = s * (1.0f / 64.0f);
    }
}

// ---------------------------------------------------------------------------
// K2: gate GEMM for batch element 0 only (the only consumed row):
//   act[p, o] = relu(bn2(sum_c x[0,c,p] * wg1[o,c])),  p in [0,3136), o in [0,32)
// accumulated over p into gsum[32].  fp32 WMMA 16x16x4, K-loop over 192.
// wg1 (24KB) staged into LDS with CDNA5 async global->LDS copies (ASYNCcnt).
// 49 blocks * 4 waves, each wave = 16 positions x 32 outputs (two 16x16 tiles).
// ---------------------------------------------------------------------------
__global__ void gate_wmma_kernel(const float* __restrict__ x,
                                 const float* __restrict__ wg1,
                                 const float* __restrict__ bn2_g, const float* __restrict__ bn2_b,
                                 const float* __restrict__ bn2_m, const float* __restrict__ bn2_v,
                                 float* __restrict__ gsum) {
    __shared__ float ldsW[32 * CCH];                 // 24 KB: wg1 staged for B-fragments
    const int tid = threadIdx.x;

    // Async stage: each of 128 threads issues 12 x 16B global->LDS copies.
    {
        unsigned ldsOff = (unsigned)(uintptr_t)(void*)ldsW;   // low 32 bits = LDS byte offset
        for (int i = tid * 4; i < 32 * CCH; i += 128 * 4) {
            asm volatile("global_load_async_to_lds_b128 %0, %1, off"
                         :: "v"(ldsOff + (unsigned)(i * 4)), "v"(wg1 + i)
                         : "memory");
        }
        asm volatile("s_wait_asynccnt 0x0" ::: "memory");
    }
    __syncthreads();

    const int wave = tid >> 5;
    const int lane = tid & 31;
    const int nn   = lane & 15;                      // column / row-in-tile index
    const int half = lane >> 4;                      // K sub-pair selector
    const int p0   = (blockIdx.x * 4 + wave) * 16;   // 16-position stripe (196 stripes total)

    // BN2 folded constants per output channel (per lane, both N-tiles)
    float sc[2], sh[2];
#pragma unroll
    for (int t = 0; t < 2; ++t) {
        int o = t * 16 + nn;
        float s = bn2_g[o] * rsqrtf(bn2_v[o] + EPSV);
        sc[t] = s;
        sh[t] = bn2_b[o] - bn2_m[o] * s;
    }

    v8f acc0 = {};
    v8f acc1 = {};
    const float* xa = x + p0 + nn;                   // n = 0 slice
    for (int kt = 0; kt < CCH / 4; ++kt) {
        const int kb = kt * 4 + 2 * half;
        // A: 16x4 fp32 fragment (ISA 7.12.2): v[0]=K(kb), v[1]=K(kb+1) for M=nn
        v2f a;
        a.x = xa[(size_t)kb * HWSZ];
        a.y = xa[(size_t)(kb + 1) * HWSZ];
        // B: 4x16 fragments from LDS, N striped across lanes, B[k][n] = wg1[n][k]
        v2f b0, b1;
        b0.x = ldsW[nn * CCH + kb];
        b0.y = ldsW[nn * CCH + kb + 1];
        b1.x = ldsW[(16 + nn) * CCH + kb];
        b1.y = ldsW[(16 + nn) * CCH + kb + 1];
        acc0 = __builtin_amdgcn_wmma_f32_16x16x4_f32(false, a, false, b0, (short)0, acc0, false, false);
        acc1 = __builtin_amdgcn_wmma_f32_16x16x4_f32(false, a, false, b1, (short)0, acc1, false, false);
    }

    // BN + ReLU on accumulator fragments, reduce over the 16 M rows.
    // Lane holds column N=nn; v8f entries are rows M = j + 8*half.
    float s0 = 0.0f, s1 = 0.0f;
#pragma unroll
    for (int j = 0; j < 8; ++j) {
        s0 += fmaxf(acc0[j] * sc[0] + sh[0], 0.0f);
        s1 += fmaxf(acc1[j] * sc[1] + sh[1], 0.0f);
    }
    s0 += __shfl_xor(s0, 16);
    s1 += __shfl_xor(s1, 16);
    if (lane < 16) {
        atomicAdd(&gsum[nn],      s0);
        atomicAdd(&gsum[16 + nn], s1);
    }
}

// ---------------------------------------------------------------------------
// K3 (single block): gate MLP for n=0, channel partition, DCT filter build.
// ---------------------------------------------------------------------------
__global__ void gate_small_kernel(const float* __restrict__ gsum,
                                  const float* __restrict__ wg2, const float* __restrict__ bg2,
                                  const float* __restrict__ beta,
                                  float* __restrict__ filt) {
    __shared__ float g0[32];
    __shared__ float xg0[32];
    __shared__ float param[33];
    const int tid = threadIdx.x;

    if (tid < 32) g0[tid] = gsum[tid] * (1.0f / 3136.0f);
    __syncthreads();
    if (tid < 32) {
        float a = bg2[tid];
        for (int k = 0; k < 32; ++k) a += g0[k] * wg2[tid * 32 + k];
        xg0[tid] = fmaxf(tanhf(a), 0.0f) + beta[0];
    }
    __syncthreads();
    if (tid == 0) {
        float s = 0.0f;
        for (int i = 0; i < 32; ++i) s += xg0[i];
        float inv = 1.0f / s;
        float psum = 0.0f;
        param[0] = 0.0f;
        for (int i = 0; i < 31; ++i) {
            float p = rintf(xg0[i] * inv * (float)CCH);   // round half-to-even, like jnp.round
            param[i + 1] = p;
            psum += p;
        }
        param[32] = (float)CCH - psum;
    }
    __syncthreads();
    for (int c = tid; c < CCH; c += blockDim.x) {
        const float fc = (float)c;
        int sel = -1;                                     // later i overwrites earlier
        for (int i = 0; i < 32; ++i)
            if (fc >= param[i] && fc < param[i + 1]) sel = i;
        for (int a = 0; a < DCTN; ++a)
            for (int b = 0; b < DCTN; ++b) {
                float v = 0.0f;
                if (sel >= 0) v = bfc(cLOWX[sel], a) * bfc(cLOWY[sel], b);
                filt[c * 49 + a * 7 + b] = v;
            }
    }
}

// ---------------------------------------------------------------------------
// K4: per-n spectral pooling + center-tap conv1d + BN1 + ReLU -> yscale = 1+y
// One block per n, one thread per channel.
// ---------------------------------------------------------------------------
__global__ void yvec_kernel(const float* __restrict__ xp, const float* __restrict__ filt,
                            const float* __restrict__ w1d, const float* __restrict__ b1d,
                            const float* __restrict__ bn1_g, const float* __restrict__ bn1_b,
                            const float* __restrict__ bn1_m, const float* __restrict__ bn1_v,
                            float* __restrict__ yscale) {
    __shared__ float yw[CCH];
    const int n = blockIdx.x;
    const int c = threadIdx.x;

    const float* xr = xp + ((size_t)n * CCH + c) * 49;
    const float* fr = filt + (size_t)c * 49;
    float a = 0.0f;
    for (int j = 0; j < 49; ++j) a += xr[j] * fr[j];
    yw[c] = a;
    __syncthreads();

    float acc = b1d[c];
    const float* wr = w1d + (size_t)c * CCH * 3 + 1;      // center tap w1d[c, k, 1]
    for (int k = 0; k < CCH; ++k) acc += yw[k] * wr[k * 3];
    float s = bn1_g[c] * rsqrtf(bn1_v[c] + EPSV);
    float v = (acc - bn1_m[c]) * s + bn1_b[c];
    yscale[(size_t)n * CCH + c] = 1.0f + fmaxf(v, 0.0f);
}

// ---------------------------------------------------------------------------
// K5: out = x * (1 + y)[n,c], float4-vectorized (784 float4 per (n,c) row).
// x re-read should hit the 192MB L2 (warmed by K1); output stores are
// non-temporal so they do not evict x from L2.
// ---------------------------------------------------------------------------
__global__ void apply_kernel(const v4f* __restrict__ x4, const float* __restrict__ yscale,
                             v4f* __restrict__ o4, int total4) {
    int idx = blockIdx.x * blockDim.x + threadIdx.x;
    if (idx >= total4) return;
    float s = yscale[idx / 784];                          // HWSZ/4 == 784
    v4f v = x4[idx] * s;
    __builtin_nontemporal_store(v, &o4[idx]);
}

// ---------------------------------------------------------------------------
extern "C" void kernel_launch(void* const* d_in, const int* in_sizes, int n_in,
                              void* d_out, int out_size, void* d_ws, size_t ws_size,
                              hipStream_t stream) {
    const float* x     = (const float*)d_in[0];
    const float* wg1   = (const float*)d_in[1];
    const float* bn2_g = (const float*)d_in[2];
    const float* bn2_b = (const float*)d_in[3];
    const float* bn2_m = (const float*)d_in[4];
    const float* bn2_v = (const float*)d_in[5];
    const float* wg2   = (const float*)d_in[6];
    const float* bg2   = (const float*)d_in[7];
    const float* beta  = (const float*)d_in[8];
    const float* w1d   = (const float*)d_in[9];
    const float* b1d   = (const float*)d_in[10];
    const float* bn1_g = (const float*)d_in[11];
    const float* bn1_b = (const float*)d_in[12];
    const float* bn1_m = (const float*)d_in[13];
    const float* bn1_v = (const float*)d_in[14];

    // workspace layout (bytes)
    char* ws = (char*)d_ws;
    float* gsum   = (float*)(ws + 0);                      // 32 f
    float* xp     = (float*)(ws + 128);                    // 64*192*49 f = 2,408,448 B
    float* filt   = (float*)(ws + 128 + 2408448);          // 192*49 f   = 37,632 B
    float* yscale = (float*)(ws + 128 + 2408448 + 37632);  // 64*192 f   = 49,152 B

    hipMemsetAsync(gsum, 0, 32 * sizeof(float), stream);

    pool_kernel<<<NBAT * CCH, 392, 0, stream>>>(x, xp);
    gate_wmma_kernel<<<49, 128, 0, stream>>>(x, wg1, bn2_g, bn2_b, bn2_m, bn2_v, gsum);
    gate_small_kernel<<<1, 256, 0, stream>>>(gsum, wg2, bg2, beta, filt);
    yvec_kernel<<<NBAT, CCH, 0, stream>>>(xp, filt, w1d, b1d, bn1_g, bn1_b, bn1_m, bn1_v, yscale);

    const int total4 = NBAT * CCH * (HWSZ / 4);            // 9,633,792
    apply_kernel<<<(total4 + 255) / 256, 256, 0, stream>>>((const v4f*)x, yscale, (v4f*)d_out, total4);

    (void)in_sizes; (void)n_in; (void)out_size; (void)ws_size;
}